// GraphEncoder_90460601189230
// MI455X (gfx1250) — compile-verified
//
#include <hip/hip_runtime.h>

typedef __attribute__((ext_vector_type(2))) float v2f;
typedef __attribute__((ext_vector_type(8))) float v8f;

// ---------------- elementwise helpers ----------------

__global__ void k_fill(float* __restrict__ p, float val, int n) {
  int i = blockIdx.x * blockDim.x + threadIdx.x;
  if (i < n) p[i] = val;
}

__global__ void k_degree(const int* __restrict__ dst, float* __restrict__ deg, int E) {
  int i = blockIdx.x * blockDim.x + threadIdx.x;
  if (i < E) atomicAdd(&deg[dst[i]], 1.0f);
}

__global__ void k_rsqrt(float* __restrict__ d, int n) {
  int i = blockIdx.x * blockDim.x + threadIdx.x;
  if (i < n) d[i] = rsqrtf(d[i]);
}

// out[i,f] = xw[i,f] * dinv[i]^2 + bias[f]   (self-loop term + bias)
__global__ void k_self_bias(const float* __restrict__ xw, const float* __restrict__ dinv,
                            const float* __restrict__ bias, float* __restrict__ out,
                            int N, int F) {
  long long i = (long long)blockIdx.x * blockDim.x + threadIdx.x;
  long long total = (long long)N * F;
  if (i < total) {
    int node = (int)(i / F);
    int f    = (int)(i % F);
    float di = dinv[node];
    out[i] = xw[i] * di * di + bias[f];
  }
}

__global__ void k_relu(float* __restrict__ p, long long n) {
  long long i = (long long)blockIdx.x * blockDim.x + threadIdx.x;
  if (i < n) p[i] = fmaxf(p[i], 0.0f);
}

// ---------------- WMMA fp32 GEMM: C[N,Fout] = A[N,K] @ B[K,Fout] ----------------
// One wave (32 lanes) per 16x16 C tile. K stepped by 4 via V_WMMA_F32_16X16X4_F32.
// ISA 7.12.2 layouts (32-bit):
//   A 16x4: lane L (row M=L&15), half h=L>>4 -> VGPR0:K=h*2, VGPR1:K=h*2+1
//   B 4x16 / C/D 16x16: row-striped across lanes; lane L -> col N=L&15,
//   C/D VGPR r -> row M = r + (L>>4)*8.
__global__ void k_gemm_wmma_f32(const float* __restrict__ A, const float* __restrict__ B,
                                float* __restrict__ C, int N, int K, int Fout) {
  int row0 = blockIdx.x * 16;
  int col0 = blockIdx.y * 16;
  int lane = threadIdx.x;        // 0..31
  int m    = lane & 15;          // row within A tile / col within B,C tile
  int h    = lane >> 4;          // half-wave select

  int arow = row0 + m;
  if (arow >= N) arow = N - 1;   // clamp loads; EXEC must stay all-ones for WMMA

  v8f acc = {};
  for (int k0 = 0; k0 < K; k0 += 4) {
    int ka = k0 + h * 2;
    v2f a, b;
    a.x = A[(long long)arow * K + ka];
    a.y = A[(long long)arow * K + ka + 1];
    b.x = B[(long long)ka       * Fout + col0 + m];
    b.y = B[(long long)(ka + 1) * Fout + col0 + m];
    // 8 args: (neg_a, A, neg_b, B, c_mod, C, reuse_a, reuse_b)
    acc = __builtin_amdgcn_wmma_f32_16x16x4_f32(false, a, false, b, (short)0, acc,
                                                false, false);
  }

#pragma unroll
  for (int r = 0; r < 8; ++r) {
    int row = row0 + r + h * 8;
    if (row < N) C[(long long)row * Fout + col0 + m] = acc[r];
  }
}

// ---------------- edge scatter: agg[dst] += dinv[src]*dinv[dst] * feat[src] ----------------
// One thread per (edge, 4-feature chunk); float4 gather, 4 global f32 atomics.
__global__ void k_scatter(const int* __restrict__ src, const int* __restrict__ dst,
                          const float* __restrict__ dinv, const float* __restrict__ feat,
                          float* __restrict__ agg, int E, int F) {
  int Fq = F >> 2;
  long long total = (long long)E * Fq;
  long long stride = (long long)gridDim.x * blockDim.x;
  for (long long idx = (long long)blockIdx.x * blockDim.x + threadIdx.x;
       idx < total; idx += stride) {
    int e = (int)(idx / Fq);
    int c = (int)(idx % Fq) << 2;
    int s = src[e];
    int d = dst[e];
    float w = dinv[s] * dinv[d];
    const float4* fv = (const float4*)(feat + (long long)s * F + c);
    float4 v = *fv;
    float* o = agg + (long long)d * F + c;
    atomicAdd(o + 0, w * v.x);
    atomicAdd(o + 1, w * v.y);
    atomicAdd(o + 2, w * v.z);
    atomicAdd(o + 3, w * v.w);
  }
}

// ---------------- launcher ----------------

extern "C" void kernel_launch(void* const* d_in, const int* in_sizes, int n_in,
                              void* d_out, int out_size, void* d_ws, size_t ws_size,
                              hipStream_t stream) {
  const float* x   = (const float*)d_in[0];
  const int*   ei  = (const int*)  d_in[1];   // [2,E] int32 (jax default x64 off)
  const float* W1  = (const float*)d_in[2];   // [128,128]
  const float* b1  = (const float*)d_in[3];   // [128]
  const float* Wmu = (const float*)d_in[4];   // [128,64]
  const float* bmu = (const float*)d_in[5];   // [64]
  const float* Wls = (const float*)d_in[6];   // [128,64]
  const float* bls = (const float*)d_in[7];   // [64]

  const int FIN = 128, FH = 128, FO = 64;
  int N = in_sizes[0] / FIN;
  int E = in_sizes[1] / 2;
  const int* src = ei;
  const int* dst = ei + E;

  float* out_mu = (float*)d_out;
  float* out_ls = out_mu + (long long)N * FO;

  // carve workspace
  char* ws = (char*)d_ws;
  auto carve = [&](size_t bytes) {
    char* p = ws;
    ws += (bytes + 255) & ~(size_t)255;
    return (float*)p;
  };
  float* dinv = carve((size_t)N * sizeof(float));
  float* xw1  = carve((size_t)N * FH * sizeof(float));
  float* hbuf = carve((size_t)N * FH * sizeof(float));
  float* hwmu = carve((size_t)N * FO * sizeof(float));
  float* hwls = carve((size_t)N * FO * sizeof(float));

  // ---- degrees: deg = 1 + count(dst); dinv = rsqrt(deg)
  k_fill  <<<(N + 255) / 256, 256, 0, stream>>>(dinv, 1.0f, N);
  k_degree<<<(E + 255) / 256, 256, 0, stream>>>(dst, dinv, E);
  k_rsqrt <<<(N + 255) / 256, 256, 0, stream>>>(dinv, N);

  int ntiles = (N + 15) / 16;

  // ---- layer 1: h = relu( scatter(norm * xw1[src]) + xw1*dinv^2 + b1 )
  {
    dim3 grid(ntiles, FH / 16);
    k_gemm_wmma_f32<<<grid, 32, 0, stream>>>(x, W1, xw1, N, FIN, FH);
    long long tot = (long long)N * FH;
    k_self_bias<<<(unsigned)((tot + 255) / 256), 256, 0, stream>>>(xw1, dinv, b1, hbuf, N, FH);
    long long work = (long long)E * (FH / 4);
    unsigned blocks = (unsigned)((work + 255) / 256);
    k_scatter<<<blocks, 256, 0, stream>>>(src, dst, dinv, xw1, hbuf, E, FH);
    k_relu<<<(unsigned)((tot + 255) / 256), 256, 0, stream>>>(hbuf, tot);
  }

  // ---- layer 2: mu / logstd
  {
    dim3 grid(ntiles, FO / 16);
    k_gemm_wmma_f32<<<grid, 32, 0, stream>>>(hbuf, Wmu, hwmu, N, FH, FO);
    k_gemm_wmma_f32<<<grid, 32, 0, stream>>>(hbuf, Wls, hwls, N, FH, FO);

    long long tot = (long long)N * FO;
    k_self_bias<<<(unsigned)((tot + 255) / 256), 256, 0, stream>>>(hwmu, dinv, bmu, out_mu, N, FO);
    k_self_bias<<<(unsigned)((tot + 255) / 256), 256, 0, stream>>>(hwls, dinv, bls, out_ls, N, FO);

    long long work = (long long)E * (FO / 4);
    unsigned blocks = (unsigned)((work + 255) / 256);
    k_scatter<<<blocks, 256, 0, stream>>>(src, dst, dinv, hwmu, out_mu, E, FO);
    k_scatter<<<blocks, 256, 0, stream>>>(src, dst, dinv, hwls, out_ls, E, FO);
  }
}